// ShortConv_41721312313465
// MI455X (gfx1250) — compile-verified
//
#include <hip/hip_runtime.h>
#include <stdint.h>

typedef __attribute__((ext_vector_type(4))) float fv4;
typedef __attribute__((ext_vector_type(4))) unsigned int u32x4;
typedef __attribute__((ext_vector_type(8))) int i32x8;
typedef __attribute__((ext_vector_type(4))) int i32x4;

#define EPS_ 1e-6f
#define B_ 4
#define S_ 4096
#define G_ 4
#define D_ 1024
#define K_ 4
#define TS_ 32
#define NROWS_ (TS_ + K_ - 1)     /* 35 rows staged in LDS */
#define ROWSTRIDE_ (G_ * D_)      /* 4096 elements between consecutive s rows */

__global__ __launch_bounds__(256)
void ShortConv_41721312313465_kernel(const float* __restrict__ x,
                                     const float* __restrict__ scale,
                                     const float* __restrict__ kern,
                                     float* __restrict__ out)
{
    __shared__ __align__(16) float lds_rows[NROWS_ * D_];   // 143,360 B
    __shared__ float lds_invr[NROWS_];

    const int tid  = threadIdx.x;
    const int sblk = blockIdx.x;
    const int g    = blockIdx.y;
    const int b    = blockIdx.z;
    const int s0   = sblk * TS_;

    // Tile = rows [s0-3, s0+TS) of the (b,g) strip. First block zero-pads 3 rows.
    const bool     first       = (s0 == 0);
    const int      s_start     = first ? 0 : (s0 - (K_ - 1));
    const int      nload       = first ? TS_ : NROWS_;
    const uint32_t lds_row_off = first ? (uint32_t)((K_ - 1) * D_ * sizeof(float)) : 0u;
    const float*   gsrc        = x + ((uint64_t)((uint64_t)b * S_ + s_start) * G_ + g) * D_;

    if (first) {
        fv4 zero = {0.f, 0.f, 0.f, 0.f};
        fv4* lp = (fv4*)lds_rows;
        for (int i = tid; i < (K_ - 1) * D_ / 4; i += 256) lp[i] = zero;
    }

#if __has_builtin(__builtin_amdgcn_tensor_load_to_lds) && __has_builtin(__builtin_amdgcn_s_wait_tensorcnt)
    if (tid < 32) {  // single wave issues the TDM descriptor (EXEC ignored by tensor ops)
        uint64_t ga    = (uint64_t)(uintptr_t)gsrc;
        uint32_t lbase = (uint32_t)(uintptr_t)lds_rows + lds_row_off;
        u32x4 g0;
        g0[0] = 1u;                                               // count=1, user D#
        g0[1] = lbase;                                            // lds_addr (bytes)
        g0[2] = (uint32_t)ga;                                     // global_addr[31:0]
        g0[3] = (uint32_t)((ga >> 32) & 0x01FFFFFFu) | (2u << 30);// global_addr[56:32] | type=2
        i32x8 g1;
        g1[0] = (int)(2u << 16);                                  // data_size=2 (4 B), no multicast/pad/iterate
        g1[1] = (int)((uint32_t)(D_ & 0xFFFF) << 16);             // tensor_dim0[15:0] @ bits 63:48
        g1[2] = (int)((((uint32_t)D_) >> 16) |
                      ((uint32_t)(nload & 0xFFFF) << 16));        // tensor_dim0 hi | tensor_dim1 lo
        g1[3] = (int)((((uint32_t)nload) >> 16) |
                      ((uint32_t)(D_ & 0xFFFF) << 16));           // tensor_dim1 hi | tile_dim0
        g1[4] = (int)(uint32_t)(nload & 0xFFFF);                  // tile_dim1 | tile_dim2=0
        g1[5] = (int)(uint32_t)ROWSTRIDE_;                        // tensor_dim0_stride[31:0] (elems)
        g1[6] = 0;                                                // stride0 hi | tensor_dim1_stride lo
        g1[7] = 0;
        i32x4 gz4 = {0, 0, 0, 0};                                 // groups 2/3 unused (2-D tensor)
        i32x8 gz8 = {0, 0, 0, 0, 0, 0, 0, 0};
        __builtin_amdgcn_tensor_load_to_lds(g0, g1, gz4, gz4, gz8, 0);
        __builtin_amdgcn_s_wait_tensorcnt(0);
    }
#else
    for (int r = 0; r < nload; ++r) {
        const fv4* src = (const fv4*)(gsrc + (uint64_t)r * ROWSTRIDE_);
        fv4* dst = (fv4*)(lds_rows + lds_row_off / 4 + r * D_);
        dst[tid] = src[tid];
    }
#endif
    __syncthreads();

    // Per-row inverse RMS: rows striped over 8 waves; wave32 butterfly reduction.
    {
        const int wave = tid >> 5, lane = tid & 31;
        for (int r = wave; r < NROWS_; r += 8) {
            const fv4* row = (const fv4*)(lds_rows + r * D_);
            float acc = 0.f;
#pragma unroll
            for (int j = 0; j < 8; ++j) {
                fv4 v = row[lane + j * 32];
                acc += v.x * v.x + v.y * v.y + v.z * v.z + v.w * v.w;
            }
#pragma unroll
            for (int off = 16; off > 0; off >>= 1)
                acc += __shfl_xor(acc, off, 32);
            if (lane == 0)
                lds_invr[r] = rsqrtf(acc * (1.0f / (float)D_) + EPS_);
        }
    }
    __syncthreads();

    // Sliding-window depthwise conv + SiLU; each thread owns a 16-byte column chunk.
    const int col   = tid * 4;
    const int cbase = g * D_ + col;
    const fv4 k0 = *(const fv4*)(kern + 0 * ROWSTRIDE_ + cbase);
    const fv4 k1 = *(const fv4*)(kern + 1 * ROWSTRIDE_ + cbase);
    const fv4 k2 = *(const fv4*)(kern + 2 * ROWSTRIDE_ + cbase);
    const fv4 k3 = *(const fv4*)(kern + 3 * ROWSTRIDE_ + cbase);
    const fv4 sc = *(const fv4*)(scale + cbase);

    const fv4* lrow = (const fv4*)lds_rows;
    fv4 w0 = lrow[0 * (D_ / 4) + tid] * lds_invr[0];
    fv4 w1 = lrow[1 * (D_ / 4) + tid] * lds_invr[1];
    fv4 w2 = lrow[2 * (D_ / 4) + tid] * lds_invr[2];

    float* obase = out + ((uint64_t)((uint64_t)b * S_ + s0) * G_ + g) * D_ + col;
#pragma unroll 4
    for (int r = 0; r < TS_; ++r) {
        fv4 v = lrow[(r + 3) * (D_ / 4) + tid] * lds_invr[r + 3];
        fv4 y = k0 * w0 + k1 * w1 + k2 * w2 + k3 * v;
        y *= sc;
        fv4 s;
        s.x = 1.0f / (1.0f + __expf(-y.x));
        s.y = 1.0f / (1.0f + __expf(-y.y));
        s.z = 1.0f / (1.0f + __expf(-y.z));
        s.w = 1.0f / (1.0f + __expf(-y.w));
        y *= s;
        *(fv4*)(obase + (uint64_t)r * ROWSTRIDE_) = y;
        w0 = w1; w1 = w2; w2 = v;
    }
}

extern "C" void kernel_launch(void* const* d_in, const int* in_sizes, int n_in,
                              void* d_out, int out_size, void* d_ws, size_t ws_size,
                              hipStream_t stream) {
    (void)in_sizes; (void)n_in; (void)out_size; (void)d_ws; (void)ws_size;
    const float* x     = (const float*)d_in[0];
    const float* scale = (const float*)d_in[1];
    const float* kern  = (const float*)d_in[2];
    float* out         = (float*)d_out;
    dim3 grid(S_ / TS_, G_, B_);
    dim3 block(256);
    ShortConv_41721312313465_kernel<<<grid, block, 0, stream>>>(x, scale, kern, out);
}